// GSL_7060926234912
// MI455X (gfx1250) — compile-verified
//
#include <hip/hip_runtime.h>

// ---------------------------------------------------------------------------
// Fused  adj = W @ W^T  +  per-row top-32  +  sort-by-index  for MI455X.
//   W: [16384, 128] fp32.  Output (floats, concatenated):
//     [0,        N*K)   : row ids (repeated)
//     [N*K,    2*N*K)   : sorted column indices
//     [2*N*K,  3*N*K)   : corresponding similarity values
// GEMM on V_WMMA_F32_16X16X4_F32 (exact fp32, wave32).
// B tiles are double-buffered in LDS and filled with
// GLOBAL_LOAD_ASYNC_TO_LDS_B128 (ASYNCcnt), fully overlapping global latency
// with the two independent WMMA accumulator chains of the current tile.
// ---------------------------------------------------------------------------

typedef float v2f __attribute__((ext_vector_type(2)));
typedef float v8f __attribute__((ext_vector_type(8)));

#define NROWS   16384
#define DIM     128
#define TOPK    32
#define KCHUNKS (DIM / 4)            // 32 k-chunks of 4
#define WAVES   4
#define THREADS (WAVES * 32)
#define ROWS_PER_BLOCK (WAVES * 16)  // 64
#define CTILE   32                   // columns per iteration (two 16-wide WMMA tiles)
#define NTILES  (NROWS / CTILE)      // 512 iterations
#define BSTRIDE (DIM + 4)            // 132 dwords: 16B-aligned rows, conflict-friendly
#define STAGE_IT ((CTILE * DIM) / (THREADS * 4))   // 8 b128 copies per thread

__global__ __launch_bounds__(THREADS)
void gsl_topk_wmma(const float* __restrict__ W, float* __restrict__ out) {
    __shared__ float Bs[2][CTILE][BSTRIDE];  // double-buffered B tile (2 x 16.9 KB)
    __shared__ float Sc[WAVES][16][33];      // per-wave 16x32 score tile (padded)

    const int tid   = threadIdx.x;
    const int lane  = tid & 31;
    const int wave  = tid >> 5;
    const int m     = lane & 15;             // row-in-tile (A) / col-in-tile (B,C)
    const int half  = lane >> 4;             // 0 or 1
    const int koff  = half << 1;             // k sub-offset: {0,1} or {2,3}
    const int rhalf = half << 3;             // C rows: vgpr r holds M = r + rhalf
    const int r0    = blockIdx.x * ROWS_PER_BLOCK + wave * 16;

    // this thread's staging slots (compile-time c/k per i)
    const int flat0 = tid * 4;

    // ---- issue async copy of tile 0 into buffer 0 (prologue)
    {
#pragma unroll
        for (int i = 0; i < STAGE_IT; ++i) {
            int flat = i * THREADS * 4 + flat0;
            int c = flat / DIM, k = flat % DIM;
            unsigned long long g = (unsigned long long)(const void*)(W + (size_t)c * DIM + k);
            unsigned int       l = (unsigned int)(unsigned long long)(const void*)&Bs[0][c][k];
            asm volatile("global_load_async_to_lds_b128 %0, %1, off"
                         :: "v"(l), "v"(g) : "memory");
        }
    }

    // ---- preload A fragments: 16 rows x 128 k (64 VGPRs/lane, reused 512x)
    v2f a[KCHUNKS];
    const float* arow = W + (size_t)(r0 + m) * DIM;
#pragma unroll
    for (int kc = 0; kc < KCHUNKS; ++kc)
        a[kc] = *(const v2f*)(arow + kc * 4 + koff);

    // ---- per-lane top-k: lane L owns row (L&15), column half (L>>4)
    float topv[TOPK];
    int   topi[TOPK];
#pragma unroll
    for (int j = 0; j < TOPK; ++j) { topv[j] = -3.4e38f; topi[j] = 0; }
    float minv = -3.4e38f;
    int   minpos = 0;

    for (int t = 0; t < NTILES; ++t) {
        const int c0  = t * CTILE;
        const int cur = t & 1;

        // our async copies for this buffer have landed; barrier covers all waves
        asm volatile("s_wait_asynccnt 0" ::: "memory");
        __syncthreads();

        // ---- issue async copy of NEXT tile into the other buffer (no wait):
        //      overlaps all of this tile's WMMA + top-k work. Branchless wrap;
        //      the final stray copy targets a dead buffer and is fenced below.
        {
            const int tn = (t + 1) & (NTILES - 1);
#pragma unroll
            for (int i = 0; i < STAGE_IT; ++i) {
                int flat = i * THREADS * 4 + flat0;
                int c = flat / DIM, k = flat % DIM;
                unsigned long long g = (unsigned long long)(const void*)
                                       (W + (size_t)(tn * CTILE + c) * DIM + k);
                unsigned int       l = (unsigned int)(unsigned long long)
                                       (const void*)&Bs[cur ^ 1][c][k];
                asm volatile("global_load_async_to_lds_b128 %0, %1, off"
                             :: "v"(l), "v"(g) : "memory");
            }
        }

        // ---- two independent 16x16 fp32 WMMA chains (cols 0-15 and 16-31)
        v8f acc0 = {0.f, 0.f, 0.f, 0.f, 0.f, 0.f, 0.f, 0.f};
        v8f acc1 = {0.f, 0.f, 0.f, 0.f, 0.f, 0.f, 0.f, 0.f};
#pragma unroll
        for (int kc = 0; kc < KCHUNKS; ++kc) {
            v2f b0 = *(const v2f*)&Bs[cur][m][kc * 4 + koff];
            v2f b1 = *(const v2f*)&Bs[cur][16 + m][kc * 4 + koff];
            acc0 = __builtin_amdgcn_wmma_f32_16x16x4_f32(
                false, a[kc], false, b0, (short)0, acc0, false, false);
            acc1 = __builtin_amdgcn_wmma_f32_16x16x4_f32(
                false, a[kc], false, b1, (short)0, acc1, false, false);
        }

        // ---- scatter both C tiles to LDS (lane holds rows rhalf..rhalf+7)
#pragma unroll
        for (int r = 0; r < 8; ++r) {
            Sc[wave][rhalf + r][m]      = acc0[r];
            Sc[wave][rhalf + r][16 + m] = acc1[r];
        }
        asm volatile("s_wait_dscnt 0" ::: "memory");   // cross-lane RAW, same wave

        // ---- streaming top-32: every lane scans 16 scores of its (row, half)
        {
            const int row   = lane & 15;
            const int cbase = half << 4;               // 0 or 16 within tile
#pragma unroll
            for (int n = 0; n < 16; ++n) {
                float v = Sc[wave][row][cbase + n];
                if (v > minv) {
                    int col = c0 + cbase + n;
#pragma unroll
                    for (int j = 0; j < TOPK; ++j)     // static-index replace
                        if (j == minpos) { topv[j] = v; topi[j] = col; }
                    minv = topv[0]; minpos = 0;        // recompute running min
#pragma unroll
                    for (int j = 1; j < TOPK; ++j)
                        if (topv[j] < minv) { minv = topv[j]; minpos = j; }
                }
            }
        }
    }

    // ---- merge half-row lists: lanes>=16 dump to (now dead) Bs, lanes<16 absorb
    asm volatile("s_wait_asynccnt 0" ::: "memory");    // stray last copy landed
    __syncthreads();                                   // all waves done with Bs
    float* dump = &Bs[0][0][0] + wave * (16 * TOPK * 2);  // per-wave 4 KB region
    if (lane >= 16) {
        float* dv = dump + (lane & 15) * (TOPK * 2);
#pragma unroll
        for (int j = 0; j < TOPK; ++j) {
            dv[2 * j]     = topv[j];
            dv[2 * j + 1] = __int_as_float(topi[j]);
        }
    }
    asm volatile("s_wait_dscnt 0" ::: "memory");       // same-wave cross-lane RAW

    if (lane < 16) {
        const float* dv = dump + lane * (TOPK * 2);
#pragma unroll
        for (int jj = 0; jj < TOPK; ++jj) {
            float v = dv[2 * jj];
            if (v > minv) {
                int col = __float_as_int(dv[2 * jj + 1]);
#pragma unroll
                for (int j = 0; j < TOPK; ++j)
                    if (j == minpos) { topv[j] = v; topi[j] = col; }
                minv = topv[0]; minpos = 0;
#pragma unroll
                for (int j = 1; j < TOPK; ++j)
                    if (topv[j] < minv) { minv = topv[j]; minpos = j; }
            }
        }

        // ---- sort the 32 survivors by column index (bitonic, static indices)
#pragma unroll
        for (int k = 2; k <= TOPK; k <<= 1) {
#pragma unroll
            for (int j = k >> 1; j > 0; j >>= 1) {
#pragma unroll
                for (int i = 0; i < TOPK; ++i) {
                    int l = i ^ j;
                    if (l > i) {
                        bool up = ((i & k) == 0);
                        bool sw = up ? (topi[i] > topi[l]) : (topi[i] < topi[l]);
                        if (sw) {
                            int   ti = topi[i]; topi[i] = topi[l]; topi[l] = ti;
                            float tv = topv[i]; topv[i] = topv[l]; topv[l] = tv;
                        }
                    }
                }
            }
        }

        const int    row = r0 + lane;
        const size_t NK  = (size_t)NROWS * TOPK;
        float* o = out + (size_t)row * TOPK;
#pragma unroll
        for (int j = 0; j < TOPK; ++j) {
            o[j]          = (float)row;       // edge_index[0]
            o[NK + j]     = (float)topi[j];   // edge_index[1] (sorted)
            o[2 * NK + j] = topv[j];          // edge_values
        }
    }
}

extern "C" void kernel_launch(void* const* d_in, const int* in_sizes, int n_in,
                              void* d_out, int out_size, void* d_ws, size_t ws_size,
                              hipStream_t stream) {
    (void)in_sizes; (void)n_in; (void)out_size; (void)d_ws; (void)ws_size;
    // inputs: d_in[0] = x (unused), d_in[1] = W [16384,128] fp32
    const float* W = (const float*)d_in[1];
    float* out = (float*)d_out;
    dim3 grid(NROWS / ROWS_PER_BLOCK);   // 256 blocks
    dim3 block(THREADS);                 // 128 threads = 4 waves
    gsl_topk_wmma<<<grid, block, 0, stream>>>(W, out);
}